// GNNModel_63247688401329
// MI455X (gfx1250) — compile-verified
//
#include <hip/hip_runtime.h>
#include <hip/hip_bf16.h>
#include <cstdint>
#include <cstddef>

// ---------------------------------------------------------------------------
// 2-layer GCN for MI455X (gfx1250, wave32).
//  * GEMMs: V_WMMA_F32_16X16X4_F32, one wave per 16-row stripe. W is staged
//    in LDS *transposed* with row stride K+4 dwords: each B operand is one
//    bank-conflict-free ds_load_b64 (stride ≡ 4 mod 64 → 32 lanes hit 64
//    distinct banks), no VGPR shuffling.
//  * ReLU is fused into the A-operand load of GEMM-2 (saves a full pass).
//  * Edge aggregation: float4 gathers + hardware global_atomic_add_f32.
//    h/agg buffers (25.6 MB) are L2-resident (192 MB L2).
// ---------------------------------------------------------------------------

typedef __attribute__((ext_vector_type(2))) float v2f;
typedef __attribute__((ext_vector_type(8))) float v8f;

__device__ __forceinline__ void atomic_add_f32(float* p, float v) {
#if defined(__HIP_PLATFORM_AMD__)
    unsafeAtomicAdd(p, v);   // hardware global_atomic_add_f32 (no CAS loop)
#else
    atomicAdd(p, v);
#endif
}

// ----------------------------- normalization -------------------------------

__global__ void k_init_deg(float* __restrict__ deg, int n) {
    int i = blockIdx.x * blockDim.x + threadIdx.x;
    if (i < n) deg[i] = 1.0f;                     // self loop contributes 1
}

__global__ void k_count_deg(const int* __restrict__ col, float* __restrict__ deg, int e) {
    int i = blockIdx.x * blockDim.x + threadIdx.x;
    if (i < e) atomic_add_f32(&deg[col[i]], 1.0f);
}

__global__ void k_rsqrt(float* __restrict__ deg, int n) {
    int i = blockIdx.x * blockDim.x + threadIdx.x;
    if (i < n) deg[i] = rsqrtf(deg[i]);           // deg >= 1, safe
}

__global__ void k_norm(const int* __restrict__ row, const int* __restrict__ col,
                       const float* __restrict__ dinv, float* __restrict__ norm,
                       int nE, int nN) {
    int e = blockIdx.x * blockDim.x + threadIdx.x;
    if (e >= nE + nN) return;
    int r, c;
    if (e < nE) { r = row[e]; c = col[e]; }
    else        { r = c = e - nE; }               // self loop
    norm[e] = dinv[r] * dinv[c];
}

// ------------------------------ WMMA GEMM ----------------------------------
// C[M x NOUT] = op(A)[M x K] * W[K x NOUT], row-major f32; op = ReLU or id.
// V_WMMA_F32_16X16X4_F32 operand mapping (ISA 7.12.2):
//   A: lane(m,khalf) holds A[m][2*khalf], A[m][2*khalf+1]  (contiguous b64)
//   B: lane(m,khalf) holds W[2*khalf][n], W[2*khalf+1][n]  -> from transposed
//      LDS sWt[n][k] (stride K+4) this is one contiguous ds_load_b64
//   C/D: vgpr r, lane(m,khalf) -> C[r + 8*khalf][n]
template <int K, int NOUT, bool RELU>
__global__ __launch_bounds__(256) void k_gemm_wmma(const float* __restrict__ A,
                                                   const float* __restrict__ W,
                                                   float* __restrict__ C, int M) {
    constexpr int KP = K + 4;                     // pad: KP ≡ 4 (mod 64), even
    __shared__ float sWt[NOUT * KP];
    for (int i = threadIdx.x; i < K * NOUT; i += blockDim.x) {
        int k = i / NOUT, n = i - k * NOUT;       // W is [K][NOUT] row-major
        sWt[n * KP + k] = W[i];
    }
    __syncthreads();

    const int lane  = threadIdx.x & 31;
    const int wave  = blockIdx.x * (256 / 32) + (threadIdx.x >> 5);
    const int row0  = wave * 16;
    if (row0 >= M) return;                        // wave-uniform: EXEC stays all-ones

    const int m16   = lane & 15;
    const int khalf = lane >> 4;                  // 0 or 1
    constexpr int NT = NOUT / 16;

    v8f acc[NT] = {};

    const float* arow = A + (size_t)(row0 + m16) * K + 2 * khalf;
#pragma unroll 4
    for (int k0 = 0; k0 < K; k0 += 4) {
        v2f a;
        a.x = arow[k0];
        a.y = arow[k0 + 1];                       // contiguous -> global b64 load
        if (RELU) { a.x = fmaxf(a.x, 0.0f); a.y = fmaxf(a.y, 0.0f); }
        const int kb = k0 + 2 * khalf;
#pragma unroll
        for (int t = 0; t < NT; ++t) {
            // aligned contiguous pair -> one ds_load_b64, conflict-free
            v2f b = *reinterpret_cast<const v2f*>(&sWt[(t * 16 + m16) * KP + kb]);
            acc[t] = __builtin_amdgcn_wmma_f32_16x16x4_f32(
                false, a, false, b, (short)0, acc[t], false, false);
        }
    }

#pragma unroll
    for (int t = 0; t < NT; ++t)
#pragma unroll
        for (int r = 0; r < 8; ++r)
            C[(size_t)(row0 + r + 8 * khalf) * NOUT + t * 16 + m16] = acc[t][r];
}

// --------------------------- bias / scatter --------------------------------

template <int F>
__global__ void k_bias_init(const float* __restrict__ b, float* __restrict__ out, int total) {
    int i = blockIdx.x * blockDim.x + threadIdx.x;
    if (i < total) out[i] = b[i & (F - 1)];
}

// F/4 lanes per edge, each handles 4 consecutive features (float4 gather,
// 4 scalar f32 atomic adds; scatter target stays in L2).
template <int F>
__global__ void k_aggregate(const float* __restrict__ h, const int* __restrict__ row,
                            const int* __restrict__ col, const float* __restrict__ norm,
                            float* __restrict__ out, int nE, int nN) {
    constexpr int LPE = F / 4;
    int t = blockIdx.x * blockDim.x + threadIdx.x;
    int e = t / LPE;
    if (e >= nE + nN) return;
    int f4 = (t - e * LPE) * 4;
    int r, c;
    if (e < nE) { r = row[e]; c = col[e]; }
    else        { r = c = e - nE; }
    float w = norm[e];
    const float4 hv = *reinterpret_cast<const float4*>(h + (size_t)r * F + f4);
    float* dst = out + (size_t)c * F + f4;
    atomic_add_f32(dst + 0, hv.x * w);
    atomic_add_f32(dst + 1, hv.y * w);
    atomic_add_f32(dst + 2, hv.z * w);
    atomic_add_f32(dst + 3, hv.w * w);
}

// ------------------------------- launcher ----------------------------------

extern "C" void kernel_launch(void* const* d_in, const int* in_sizes, int n_in,
                              void* d_out, int out_size, void* d_ws, size_t ws_size,
                              hipStream_t stream) {
    constexpr int IN_DIM = 128, HID = 64, OUT = 32;

    const float* x  = (const float*)d_in[0];
    const int*   ei = (const int*)d_in[1];
    const float* W1 = (const float*)d_in[2];
    const float* b1 = (const float*)d_in[3];
    const float* W2 = (const float*)d_in[4];
    const float* b2 = (const float*)d_in[5];
    float* out = (float*)d_out;

    const int N_ = in_sizes[0] / IN_DIM;   // 100000
    const int E_ = in_sizes[1] / 2;        // 1600000
    const int* row = ei;                   // edge_index[0] = source
    const int* col = ei + E_;              // edge_index[1] = target

    // workspace carve-out
    char* ws = (char*)d_ws;
    size_t off = 0;
    auto carve = [&](size_t bytes) {
        char* p = ws + off;
        off += (bytes + 255) & ~(size_t)255;
        return p;
    };
    float* deg  = (float*)carve((size_t)N_ * 4);             // deg -> dinv in place
    float* norm = (float*)carve((size_t)(E_ + N_) * 4);
    float* h    = (float*)carve((size_t)N_ * HID * 4);
    float* agg1 = (float*)carve((size_t)N_ * HID * 4);
    float* h2   = (float*)carve((size_t)N_ * OUT * 4);

    auto cdiv = [](long long a, long long b) { return (int)((a + b - 1) / b); };
    dim3 blk(256);

    // normalization coefficients
    k_init_deg <<<cdiv(N_, 256), blk, 0, stream>>>(deg, N_);
    k_count_deg<<<cdiv(E_, 256), blk, 0, stream>>>(col, deg, E_);
    k_rsqrt    <<<cdiv(N_, 256), blk, 0, stream>>>(deg, N_);
    k_norm     <<<cdiv((long long)E_ + N_, 256), blk, 0, stream>>>(row, col, deg, norm, E_, N_);

    // layer 1: h = x @ W1 ; agg1 = scatter(h*norm) + b1
    k_gemm_wmma<IN_DIM, HID, false><<<cdiv(N_, 128), blk, 0, stream>>>(x, W1, h, N_);
    k_bias_init<HID><<<cdiv((long long)N_ * HID, 256), blk, 0, stream>>>(b1, agg1, N_ * HID);
    k_aggregate<HID><<<cdiv((long long)(E_ + N_) * (HID / 4), 256), blk, 0, stream>>>(
        h, row, col, norm, agg1, E_, N_);

    // layer 2: h2 = relu(agg1) @ W2 (ReLU fused) ; out = scatter(h2*norm) + b2
    k_gemm_wmma<HID, OUT, true><<<cdiv(N_, 128), blk, 0, stream>>>(agg1, W2, h2, N_);
    k_bias_init<OUT><<<cdiv((long long)N_ * OUT, 256), blk, 0, stream>>>(b2, out, N_ * OUT);
    k_aggregate<OUT><<<cdiv((long long)(E_ + N_) * (OUT / 4), 256), blk, 0, stream>>>(
        h2, row, col, norm, out, E_, N_);
}